// GCNVQAModel_33122787786760
// MI455X (gfx1250) — compile-verified
//
#include <hip/hip_runtime.h>
#include <hip/hip_bf16.h>

// ---------------------------------------------------------------------------
// GCN-VQA forward for MI455X (gfx1250, wave32).
// GEMMs use V_WMMA_F32_16X16X4_F32 (fp32 matrix pipe); edge aggregation is
// HBM-bound scatter-add with global_atomic_add_f32.
// ---------------------------------------------------------------------------

typedef __attribute__((ext_vector_type(2))) float v2f;
typedef __attribute__((ext_vector_type(8))) float v8f;

#define B_    64
#define NPG_  4096
#define N_    (B_ * NPG_)      // 262144
#define EPG_  32768
#define E_    (B_ * EPG_)      // 2097152
#define D_    64
#define CLIP_ 512

__device__ __forceinline__ v8f wmma4(v2f a, v2f b, v8f c) {
  // D = A(16x4 f32) * B(4x16 f32) + C(16x16 f32)
  return __builtin_amdgcn_wmma_f32_16x16x4_f32(
      /*neg_a=*/false, a, /*neg_b=*/false, b,
      /*c_mod=*/(short)0, c, /*reuse_a=*/false, /*reuse_b=*/false);
}

// --------------------------------------------------------------------------
// 1) img = image_feats @ Wi + bi ; txt = question_feats @ Wt + bt
//    [64,512] @ [512,64]. One wave per 16x16 output tile; 32 tiles total.
// --------------------------------------------------------------------------
__global__ __launch_bounds__(32)
void proj_kernel(const float* __restrict__ imgF, const float* __restrict__ txtF,
                 const float* __restrict__ Wi,   const float* __restrict__ bi,
                 const float* __restrict__ Wt,   const float* __restrict__ bt,
                 float* __restrict__ img_o, float* __restrict__ txt_o) {
  const int blk   = blockIdx.x;          // 0..31
  const bool isT  = blk >= 16;
  const int tile  = blk & 15;
  const int mt    = (tile >> 2) * 16;
  const int nt    = (tile & 3) * 16;
  const float* A    = isT ? txtF : imgF;
  const float* W    = isT ? Wt   : Wi;
  const float* bias = isT ? bt   : bi;
  float* O          = isT ? txt_o : img_o;

  const int l  = threadIdx.x;
  const int m  = l & 15;        // row (A) / col (B,C) within tile
  const int hi = l >> 4;        // half-wave selects K pair (ISA 16x4 f32 layout)

  v8f c = {};
  for (int k = 0; k < CLIP_; k += 4) {
    const int kk = k + hi * 2;
    v2f a; a.x = A[(mt + m) * CLIP_ + kk];
           a.y = A[(mt + m) * CLIP_ + kk + 1];
    v2f b; b.x = W[kk * D_ + nt + m];
           b.y = W[(kk + 1) * D_ + nt + m];
    c = wmma4(a, b, c);
  }
  const float bv = bias[nt + m];
  for (int v = 0; v < 8; ++v) {
    const int row = mt + v + hi * 8;     // C layout: VGPR v -> rows v / v+8
    O[row * D_ + nt + m] = c[v] + bv;
  }
}

// --------------------------------------------------------------------------
// 2) xw = (relu(X_fixed[node_idx] @ Wnp + bnp)) @ Wg
//    256-thread blocks (8 waves). Both weight matrices staged in LDS once
//    per block; each wave runs 8 node-tiles (block covers 1024 nodes).
// --------------------------------------------------------------------------
#define NODES_PER_BLOCK 1024
#define TILES_PER_WAVE  8

__global__ __launch_bounds__(256)
void node_kernel(const float* __restrict__ Xf, const int* __restrict__ nidx,
                 const float* __restrict__ Wnp, const float* __restrict__ bnp,
                 const float* __restrict__ Wg, float* __restrict__ xw) {
  __shared__ float W1[D_ * D_];          // 16 KB : Wnp
  __shared__ float W2[D_ * D_];          // 16 KB : Wg
  __shared__ float At[8][16 * D_];       // 32 KB : per-wave A tiles

  const int t    = threadIdx.x;
  const int wave = t >> 5;               // 0..7
  const int l    = t & 31;
  const int m    = l & 15;
  const int hi   = l >> 4;

  // Cooperative staging of both weight matrices (1024 float4 each)
  for (int i = 0; i < 4; ++i) {
    const int p = t + 256 * i;
    *(float4*)(W1 + p * 4) = *(const float4*)(Wnp + p * 4);
  }
  for (int i = 0; i < 4; ++i) {
    const int p = t + 256 * i;
    *(float4*)(W2 + p * 4) = *(const float4*)(Wg + p * 4);
  }
  __syncthreads();

  float* A = At[wave];
  const float b0 = bnp[ 0 + m], b1 = bnp[16 + m], b2 = bnp[32 + m], b3 = bnp[48 + m];
  const int blockBase = blockIdx.x * NODES_PER_BLOCK;

  for (int tile = 0; tile < TILES_PER_WAVE; ++tile) {
    const int row0 = blockBase + wave * (16 * TILES_PER_WAVE) + tile * 16;

    // Gather 16 rows of X_fixed (16 x 16 float4 = 256 xfers, 8 per lane)
    for (int i = 0; i < 8; ++i) {
      const int p  = l + 32 * i;
      const int r  = p >> 4;
      const int c4 = p & 15;
      const int idx = nidx[row0 + r];
      *(float4*)(A + r * D_ + c4 * 4) =
          *(const float4*)(Xf + (size_t)idx * D_ + c4 * 4);
    }
    __syncthreads();

    // GEMM1: x = relu(A @ Wnp + bnp)
    v8f c0 = {}, c1 = {}, c2 = {}, c3 = {};
    for (int k = 0; k < D_; k += 4) {
      const int kk = k + hi * 2;
      v2f a; a.x = A[m * D_ + kk]; a.y = A[m * D_ + kk + 1];
      v2f b;
      b.x = W1[kk * D_ +  0 + m]; b.y = W1[(kk + 1) * D_ +  0 + m]; c0 = wmma4(a, b, c0);
      b.x = W1[kk * D_ + 16 + m]; b.y = W1[(kk + 1) * D_ + 16 + m]; c1 = wmma4(a, b, c1);
      b.x = W1[kk * D_ + 32 + m]; b.y = W1[(kk + 1) * D_ + 32 + m]; c2 = wmma4(a, b, c2);
      b.x = W1[kk * D_ + 48 + m]; b.y = W1[(kk + 1) * D_ + 48 + m]; c3 = wmma4(a, b, c3);
    }
    __syncthreads();

    // bias + relu; restage x tile in LDS (A layout for second GEMM)
    for (int v = 0; v < 8; ++v) {
      const int r = v + hi * 8;          // C layout: VGPR v -> rows v / v+8
      float x;
      x = c0[v] + b0; A[r * D_ +  0 + m] = x > 0.f ? x : 0.f;
      x = c1[v] + b1; A[r * D_ + 16 + m] = x > 0.f ? x : 0.f;
      x = c2[v] + b2; A[r * D_ + 32 + m] = x > 0.f ? x : 0.f;
      x = c3[v] + b3; A[r * D_ + 48 + m] = x > 0.f ? x : 0.f;
    }
    __syncthreads();

    // GEMM2: xw = x @ Wg
    v8f z = {};
    c0 = z; c1 = z; c2 = z; c3 = z;
    for (int k = 0; k < D_; k += 4) {
      const int kk = k + hi * 2;
      v2f a; a.x = A[m * D_ + kk]; a.y = A[m * D_ + kk + 1];
      v2f b;
      b.x = W2[kk * D_ +  0 + m]; b.y = W2[(kk + 1) * D_ +  0 + m]; c0 = wmma4(a, b, c0);
      b.x = W2[kk * D_ + 16 + m]; b.y = W2[(kk + 1) * D_ + 16 + m]; c1 = wmma4(a, b, c1);
      b.x = W2[kk * D_ + 32 + m]; b.y = W2[(kk + 1) * D_ + 32 + m]; c2 = wmma4(a, b, c2);
      b.x = W2[kk * D_ + 48 + m]; b.y = W2[(kk + 1) * D_ + 48 + m]; c3 = wmma4(a, b, c3);
    }
    for (int v = 0; v < 8; ++v) {
      const size_t row = (size_t)(row0 + v + hi * 8) * D_;
      xw[row +  0 + m] = c0[v];
      xw[row + 16 + m] = c1[v];
      xw[row + 32 + m] = c2[v];
      xw[row + 48 + m] = c3[v];
    }
    __syncthreads();                     // A reused next iteration
  }
}

// --------------------------------------------------------------------------
// 3) degree / dinv / acc init / edge scatter (HBM-bound part)
// --------------------------------------------------------------------------
__global__ __launch_bounds__(256)
void zero_kernel(float* __restrict__ p, int n) {
  const int i = blockIdx.x * 256 + threadIdx.x;
  if (i < n) p[i] = 0.f;
}

__global__ __launch_bounds__(256)
void deg_kernel(const int* __restrict__ dst, float* __restrict__ deg) {
  const int e = blockIdx.x * 256 + threadIdx.x;
  if (e < E_) atomicAdd(&deg[dst[e]], 1.0f);
}

__global__ __launch_bounds__(256)
void dinv_kernel(float* __restrict__ deg) {
  const int n = blockIdx.x * 256 + threadIdx.x;
  if (n < N_) deg[n] = rsqrtf(deg[n] + 1.0f);   // +1 self loop
}

// acc[n][:] = xw[n][:] * dinv[n]^2   (self-loop term pre-folded)
__global__ __launch_bounds__(256)
void accinit_kernel(const float* __restrict__ xw, const float* __restrict__ dinv,
                    float* __restrict__ acc) {
  const int t  = blockIdx.x * 256 + threadIdx.x;   // N*16 threads
  const int n  = t >> 4;
  const int c4 = t & 15;
  const float di = dinv[n];
  const float s  = di * di;
  float4 v = *(const float4*)(xw + (size_t)n * D_ + c4 * 4);
  v.x *= s; v.y *= s; v.z *= s; v.w *= s;
  *(float4*)(acc + (size_t)n * D_ + c4 * 4) = v;
}

// acc[dst][:] += xw[src][:] * dinv[src]*dinv[dst]
__global__ __launch_bounds__(256)
void edge_kernel(const int* __restrict__ src, const int* __restrict__ dst,
                 const float* __restrict__ dinv, const float* __restrict__ xw,
                 float* __restrict__ acc) {
  const long long t = (long long)blockIdx.x * 256 + threadIdx.x;  // E*16 threads
  const int e  = (int)(t >> 4);
  const int c4 = (int)(t & 15);
  const int s  = src[e];
  const int d  = dst[e];
  const float coef = dinv[s] * dinv[d];
  const float4 v = *(const float4*)(xw + (size_t)s * D_ + c4 * 4);
  float* ap = acc + (size_t)d * D_ + c4 * 4;
  atomicAdd(ap + 0, v.x * coef);
  atomicAdd(ap + 1, v.y * coef);
  atomicAdd(ap + 2, v.z * coef);
  atomicAdd(ap + 3, v.w * coef);
}

// --------------------------------------------------------------------------
// 4) h = relu(acc + bg); global mean pool. Block b covers 1024 nodes of
//    graph b/4 (NPG=4096). Partial sums reduced in LDS, one atomic per dim.
// --------------------------------------------------------------------------
__global__ __launch_bounds__(256)
void pool_kernel(const float* __restrict__ acc, const float* __restrict__ bg,
                 float* __restrict__ gsum) {
  __shared__ float red[256];
  const int blk  = blockIdx.x;           // 256 blocks
  const int g    = blk >> 2;
  const int base = blk * 1024;
  const int t  = threadIdx.x;
  const int d  = t & 63;
  const int rg = t >> 6;
  const float bgv = bg[d];
  float s = 0.f;
  for (int r = rg; r < 1024; r += 4) {
    const float h = acc[(size_t)(base + r) * D_ + d] + bgv;
    s += h > 0.f ? h : 0.f;
  }
  red[t] = s;
  __syncthreads();
  if (rg == 0) {
    const float tot = red[d] + red[64 + d] + red[128 + d] + red[192 + d];
    atomicAdd(&gsum[g * D_ + d], tot);
  }
}

// --------------------------------------------------------------------------
// 5) classifier head + BCE loss (tiny): out[0..63]=logits, out[64]=loss
// --------------------------------------------------------------------------
__global__ __launch_bounds__(256)
void head_kernel(const float* __restrict__ img, const float* __restrict__ txt,
                 const float* __restrict__ gsum,
                 const float* __restrict__ Wc1, const float* __restrict__ bc1,
                 const float* __restrict__ Wc2, const float* __restrict__ bc2,
                 const float* __restrict__ answer, float* __restrict__ out) {
  __shared__ float h1[64 * 64];
  __shared__ float lg[64];
  const int t = threadIdx.x;
  const float inv = 1.0f / (float)NPG_;
  for (int o = t; o < 64 * 64; o += 256) {
    const int b = o >> 6, j = o & 63;
    float s = bc1[j];
    for (int k = 0; k < 64; ++k) s += img[b * 64 + k] * Wc1[k * 64 + j];
    for (int k = 0; k < 64; ++k) s += txt[b * 64 + k] * Wc1[(64 + k) * 64 + j];
    for (int k = 0; k < 64; ++k) s += (gsum[b * 64 + k] * inv) * Wc1[(128 + k) * 64 + j];
    h1[o] = s > 0.f ? s : 0.f;
  }
  __syncthreads();
  if (t < 64) {
    float s = bc2[0];
    for (int j = 0; j < 64; ++j) s += h1[t * 64 + j] * Wc2[j];
    lg[t] = s;
    out[t] = s;
  }
  __syncthreads();
  if (t == 0) {
    float L = 0.f;
    for (int b = 0; b < 64; ++b) {
      const float x = lg[b];
      const float a = answer[b];
      const float mx = x > 0.f ? x : 0.f;
      L += mx - x * a + log1pf(expf(-fabsf(x)));
    }
    out[64] = L / 64.0f;
  }
}

// --------------------------------------------------------------------------
extern "C" void kernel_launch(void* const* d_in, const int* in_sizes, int n_in,
                              void* d_out, int out_size, void* d_ws, size_t ws_size,
                              hipStream_t stream) {
  (void)in_sizes; (void)n_in; (void)out_size; (void)ws_size;

  const float* image_feats    = (const float*)d_in[0];
  const float* question_feats = (const float*)d_in[1];
  const float* X_fixed        = (const float*)d_in[2];
  const int*   node_idx       = (const int*)  d_in[3];
  const int*   src            = (const int*)  d_in[4];
  const int*   dst            = (const int*)  d_in[5];
  /* graph_ids d_in[6] unused: layout is implicit (n / NPG) */
  const float* answer         = (const float*)d_in[7];
  const float* Wi  = (const float*)d_in[8];
  const float* bi  = (const float*)d_in[9];
  const float* Wt  = (const float*)d_in[10];
  const float* bt  = (const float*)d_in[11];
  const float* Wnp = (const float*)d_in[12];
  const float* bnp = (const float*)d_in[13];
  const float* Wg  = (const float*)d_in[14];
  const float* bg  = (const float*)d_in[15];
  const float* Wc1 = (const float*)d_in[16];
  const float* bc1 = (const float*)d_in[17];
  const float* Wc2 = (const float*)d_in[18];
  const float* bc2 = (const float*)d_in[19];
  float* out = (float*)d_out;

  float* ws   = (float*)d_ws;
  float* xw   = ws;                         // N*64
  float* acc  = xw   + (size_t)N_ * D_;     // N*64
  float* dinv = acc  + (size_t)N_ * D_;     // N   (also used as deg)
  float* img  = dinv + N_;                  // 64*64
  float* txt  = img  + 64 * 64;             // 64*64
  float* gsum = txt  + 64 * 64;             // 64*64

  // 1) image/text projections (WMMA f32)
  proj_kernel<<<32, 32, 0, stream>>>(image_feats, question_feats,
                                     Wi, bi, Wt, bt, img, txt);

  // 2) node projection + GCN weight GEMM (fused, WMMA f32)
  node_kernel<<<N_ / NODES_PER_BLOCK, 256, 0, stream>>>(X_fixed, node_idx,
                                                        Wnp, bnp, Wg, xw);

  // 3) degrees -> dinv
  zero_kernel<<<(N_ + 255) / 256, 256, 0, stream>>>(dinv, N_);
  deg_kernel<<<E_ / 256, 256, 0, stream>>>(dst, dinv);
  dinv_kernel<<<N_ / 256, 256, 0, stream>>>(dinv);

  // 4) acc = xw * dinv^2 ; acc[dst] += coef * xw[src]
  accinit_kernel<<<(N_ * 16) / 256, 256, 0, stream>>>(xw, dinv, acc);
  edge_kernel<<<(size_t)E_ * 16 / 256, 256, 0, stream>>>(src, dst, dinv, xw, acc);

  // 5) relu + mean pool per graph
  zero_kernel<<<(B_ * D_ + 255) / 256, 256, 0, stream>>>(gsum, B_ * D_);
  pool_kernel<<<256, 256, 0, stream>>>(acc, bg, gsum);

  // 6) head + loss
  head_kernel<<<1, 256, 0, stream>>>(img, txt, gsum, Wc1, bc1, Wc2, bc2,
                                     answer, out);
}